// IdentityEmbedding_14147622273767
// MI455X (gfx1250) — compile-verified
//
#include <hip/hip_runtime.h>

// IdentityEmbedding: out[row, :] = projection[idx[row], :]
//   rows = B*T = 16384, N_EMBD = 1024 (4 KB per row, f32)
// Pure bandwidth kernel (~134 MB total -> ~5.8 us floor at 23.3 TB/s HBM).
//
// Data movement uses the gfx1250 async global<->LDS DMA path
// (GLOBAL_LOAD_ASYNC_TO_LDS_B128 / GLOBAL_STORE_ASYNC_FROM_LDS_B128, tracked
// by ASYNCcnt). Each block stages 8 rows (32 KB LDS) and each lane issues all
// 8 async loads before one s_wait_asynccnt 0, giving 4 KB outstanding per
// wave (8x the MLP of a single-row version) to cover the HBM latency-BW
// product, then drains with 8 async stores. Lanes only re-read LDS bytes
// they loaded themselves, so no workgroup barrier is needed.

#define N_EMBD  1024
#define THREADS 256   // 8 wave32s; each lane moves one float4 (16 B) per row
#define RPB     8     // rows per block -> 32 KB LDS staging

__global__ __launch_bounds__(THREADS) void IdentityEmbedding_gather_async(
    const float* __restrict__ proj,   // (VOCAB, N_EMBD) f32
    const int*   __restrict__ idx,    // (rows,) i32
    float*       __restrict__ out,    // (rows, N_EMBD) f32
    int rows)
{
    __shared__ float smem[RPB * N_EMBD];   // 8 x 4 KB staging tiles

    const int      row0 = blockIdx.x * RPB;
    const unsigned tid  = threadIdx.x;

    // Generic pointers into LDS are {aperture_hi32, lds_offset}; truncating
    // yields the wave-relative LDS byte offset the async ops' VGPR expects.
    const unsigned ldsBase = (unsigned)(size_t)(&smem[0]);

    unsigned long long src[RPB];
    unsigned long long dst[RPB];
    unsigned           lds[RPB];

#pragma unroll
    for (int r = 0; r < RPB; ++r) {
        int rr = row0 + r;
        if (rr > rows - 1) rr = rows - 1;          // benign tail clamp
        const unsigned token = (unsigned)idx[rr];  // uniform -> scalar load
        src[r] = (unsigned long long)(proj)
               + (unsigned long long)token * (N_EMBD * 4ull)
               + (unsigned long long)tid * 16ull;
        dst[r] = (unsigned long long)(out)
               + (unsigned long long)rr * (N_EMBD * 4ull)
               + (unsigned long long)tid * 16ull;
        lds[r] = ldsBase + (unsigned)r * (N_EMBD * 4u) + tid * 16u;
    }

    // Issue all 8 async loads (ASYNCcnt += 8): 4 KB in flight per wave.
#pragma unroll
    for (int r = 0; r < RPB; ++r) {
        asm volatile("global_load_async_to_lds_b128 %0, %1, off"
                     :
                     : "v"(lds[r]), "v"(src[r])
                     : "memory");
    }

    // Single wait for all loads to land in LDS (per-wave counter; each lane
    // only re-reads its own LDS bytes, so no cross-wave barrier needed).
#if __has_builtin(__builtin_amdgcn_s_wait_asynccnt)
    __builtin_amdgcn_s_wait_asynccnt(0);
#else
    asm volatile("s_wait_asynccnt 0" ::: "memory");
#endif

    // Drain: 8 async stores LDS -> output rows. S_ENDPGM performs an implicit
    // wait-idle, so stores complete before the wave retires.
#pragma unroll
    for (int r = 0; r < RPB; ++r) {
        asm volatile("global_store_async_from_lds_b128 %0, %1, off"
                     :
                     : "v"(dst[r]), "v"(lds[r])
                     : "memory");
    }
}

extern "C" void kernel_launch(void* const* d_in, const int* in_sizes, int n_in,
                              void* d_out, int out_size, void* d_ws, size_t ws_size,
                              hipStream_t stream)
{
    // setup_inputs() dict order: {"idx", "projection"}
    const int*   idx  = (const int*)d_in[0];     // B*T int32 tokens
    const float* proj = (const float*)d_in[1];   // VOCAB x N_EMBD f32
    float*       out  = (float*)d_out;           // B*T x N_EMBD f32

    const int rows    = in_sizes[0];             // B*T = 16384
    const int nblocks = (rows + RPB - 1) / RPB;  // 2048

    dim3 grid((unsigned)nblocks);
    dim3 block(THREADS);
    hipLaunchKernelGGL(IdentityEmbedding_gather_async, grid, block, 0, stream,
                       proj, idx, out, rows);
}